// SimpleHGN_18013092839753
// MI455X (gfx1250) — compile-verified
//
#include <hip/hip_runtime.h>
#include <hip/hip_bf16.h>
#include <math.h>

typedef __attribute__((ext_vector_type(2))) float v2f;
typedef __attribute__((ext_vector_type(8))) float v8f;

#define HID 128
#define LDT 68          // LDS pitch in floats: 16B-aligned rows, conflict-free frag reads
#define RELD 200

// ---------------- ordered-uint float max helpers (for segment max) ----------------
__device__ __forceinline__ unsigned ordKey(float f) {
  unsigned u = __float_as_uint(f);
  return (u & 0x80000000u) ? ~u : (u | 0x80000000u);
}
__device__ __forceinline__ float ordVal(unsigned u) {
  unsigned b = (u & 0x80000000u) ? (u ^ 0x80000000u) : ~u;
  return __uint_as_float(b);
}

// ---------------- WMMA GEMM: C[M,128] = act(A[M,K] @ W[128,K]^T + bias) ----------------
// block = 256 threads (8 wave32); block tile = 32(M) x 128(N); wave w owns N-tile w (16 cols)
// and both 16-row M subtiles (reuses its B fragment twice per wmma step).
__global__ __launch_bounds__(256)
void gemm_wmma_kernel(const float* __restrict__ A, const float* __restrict__ W,
                      const float* __restrict__ bias, float* __restrict__ C,
                      int M, int K, int act)
{
  __shared__ float ldsA[32 * LDT];
  __shared__ float ldsW[128 * LDT];

  const int tid  = threadIdx.x;
  const int lane = tid & 31;
  const int wave = tid >> 5;           // 0..7 -> N tile
  const int l15  = lane & 15;
  const int half = lane >> 4;          // 0: K(A)/M(C) low half, 1: high half
  const int m0   = blockIdx.x * 32;

  v8f acc0 = {};
  v8f acc1 = {};

  for (int kc = 0; kc < K; kc += 64) {
    // stage A tile: 32 rows x 64 cols (512 float4, 2 per thread), zero-padded past M
    #pragma unroll
    for (int i = 0; i < 2; ++i) {
      int idx = tid + i * 256;
      int r = idx >> 4, c4 = idx & 15;
      int gr = m0 + r;
      float4 v = make_float4(0.f, 0.f, 0.f, 0.f);
      if (gr < M) v = *(const float4*)(A + (size_t)gr * K + kc + c4 * 4);
      *(float4*)(&ldsA[r * LDT + c4 * 4]) = v;
    }
    // stage W chunk: 128 rows x 64 cols (2048 float4, 8 per thread)
    #pragma unroll
    for (int i = 0; i < 8; ++i) {
      int idx = tid + i * 256;
      int r = idx >> 4, c4 = idx & 15;
      float4 v = *(const float4*)(W + (size_t)r * K + kc + c4 * 4);
      *(float4*)(&ldsW[r * LDT + c4 * 4]) = v;
    }
    __syncthreads();

    if (kc + 64 < K)   // global_prefetch_b8 of next W chunk
      __builtin_prefetch(W + (size_t)(tid & 127) * K + kc + 64, 0, 3);

    #pragma unroll
    for (int ks = 0; ks < 16; ++ks) {
      const int kk = ks * 4 + 2 * half;
      // B frag (4x16): W[n][k] streamed per-lane; A frags per ISA 16x4 layout
      v2f b  = *(const v2f*)(&ldsW[(wave * 16 + l15) * LDT + kk]);
      v2f a0 = *(const v2f*)(&ldsA[(l15)      * LDT + kk]);
      v2f a1 = *(const v2f*)(&ldsA[(16 + l15) * LDT + kk]);
      acc0 = __builtin_amdgcn_wmma_f32_16x16x4_f32(false, a0, false, b, (short)0, acc0, false, false);
      acc1 = __builtin_amdgcn_wmma_f32_16x16x4_f32(false, a1, false, b, (short)0, acc1, false, false);
    }
    __syncthreads();
  }

  // epilogue: C/D layout -> VGPR r: lanes 0-15 = row r, lanes 16-31 = row r+8
  const int colg = wave * 16 + l15;
  const float bv = bias ? bias[colg] : 0.0f;
  #pragma unroll
  for (int r = 0; r < 8; ++r) {
    {
      int gr = m0 + r + 8 * half;
      float v = acc0[r] + bv;
      if (act) v = v > 0.f ? v : 0.01f * v;
      if (gr < M) C[(size_t)gr * HID + colg] = v;
    }
    {
      int gr = m0 + 16 + r + 8 * half;
      float v = acc1[r] + bv;
      if (act) v = v > 0.f ? v : 0.01f * v;
      if (gr < M) C[(size_t)gr * HID + colg] = v;
    }
  }
}

// ---------------- per-node attention scalars: sA[n]=Wx[n].aA, sB[n]=Wx[n].aB ----------------
__global__ __launch_bounds__(256)
void node_dots_kernel(const float* __restrict__ Wx, const float* __restrict__ a,
                      float* __restrict__ sA, float* __restrict__ sB, int M)
{
  int node = blockIdx.x * 8 + (threadIdx.x >> 5);
  int lane = threadIdx.x & 31;
  if (node >= M) return;
  const float* row = Wx + (size_t)node * HID;
  float pa = 0.f, pb = 0.f;
  #pragma unroll
  for (int i = 0; i < 4; ++i) {
    float v = row[lane + 32 * i];
    pa += v * a[lane + 32 * i];
    pb += v * a[HID + lane + 32 * i];
  }
  #pragma unroll
  for (int off = 16; off; off >>= 1) {
    pa += __shfl_xor(pa, off, 32);
    pb += __shfl_xor(pb, off, 32);
  }
  if (lane == 0) { sA[node] = pa; sB[node] = pb; }
}

// ---------------- sC[r] = rel[r] . (Wr^T aC) : collapses the 600k x 200 x 128 GEMM ----------------
__global__ void relscore_kernel(const float* __restrict__ Wr, const float* __restrict__ aC,
                                const float* __restrict__ rel, float* __restrict__ sC, int nRel)
{
  __shared__ float t[RELD];
  int d = threadIdx.x;
  if (d < RELD) {
    float acc = 0.f;
    for (int h = 0; h < HID; ++h) acc += Wr[(size_t)h * RELD + d] * aC[h];
    t[d] = acc;
  }
  __syncthreads();
  if (d < nRel) {
    float acc = 0.f;
    for (int i = 0; i < RELD; ++i) acc += rel[(size_t)d * RELD + i] * t[i];
    sC[d] = acc;
  }
}

// ---------------- zero agg, reset segment-max / segment-sum ----------------
__global__ void init_kernel(float* __restrict__ agg, unsigned* __restrict__ m,
                            float* __restrict__ denom, int M)
{
  size_t idx = (size_t)blockIdx.x * 256 + threadIdx.x;
  if (idx < (size_t)M * HID) agg[idx] = 0.f;
  if (idx < (size_t)M) { m[idx] = 0u; denom[idx] = 0.f; }   // 0 < ordKey(any finite)
}

// ---------------- edge score + segment max ----------------
__global__ void edge_score_kernel(const int* __restrict__ src, const int* __restrict__ dst,
                                  const int* __restrict__ et,
                                  const float* __restrict__ sA, const float* __restrict__ sB,
                                  const float* __restrict__ sC,
                                  float* __restrict__ score, unsigned* __restrict__ m, int E)
{
  int e = blockIdx.x * 256 + threadIdx.x;
  if (e >= E) return;
  int d = dst[e];
  float s = sA[d] + sB[src[e]] + sC[et[e]];
  s = s > 0.f ? s : 0.2f * s;                  // leaky_relu(., 0.2)
  score[e] = s;
  atomicMax(&m[d], ordKey(s));
}

// ---------------- edge exp + segment sum ----------------
__global__ void edge_exp_kernel(const int* __restrict__ dst, const float* __restrict__ score,
                                const unsigned* __restrict__ m,
                                float* __restrict__ ee, float* __restrict__ denom, int E)
{
  int e = blockIdx.x * 256 + threadIdx.x;
  if (e >= E) return;
  int d = dst[e];
  float ex = __expf(score[e] - ordVal(m[d]));
  ee[e] = ex;
  atomicAdd(&denom[d], ex);
}

// ---------------- scatter: agg[dst] += alpha * Wx[src]; one wave32 per edge ----------------
__global__ __launch_bounds__(256)
void edge_scatter_kernel(const int* __restrict__ src, const int* __restrict__ dst,
                         const float* __restrict__ Wx, const float* __restrict__ ee,
                         const float* __restrict__ denom, float* __restrict__ agg, int E)
{
  int e = blockIdx.x * 8 + (threadIdx.x >> 5);
  int lane = threadIdx.x & 31;
  if (e >= E) return;
  int sN = src[e], dN = dst[e];
  float alpha = ee[e] / denom[dN];
  float4 v = ((const float4*)(Wx + (size_t)sN * HID))[lane];
  float* ag = agg + (size_t)dN * HID + lane * 4;
  atomicAdd(ag + 0, alpha * v.x);
  atomicAdd(ag + 1, alpha * v.y);
  atomicAdd(ag + 2, alpha * v.z);
  atomicAdd(ag + 3, alpha * v.w);
}

// ---------------- out = elu(agg + res), optional row L2-normalize ----------------
__global__ __launch_bounds__(256)
void node_out_kernel(const float* __restrict__ agg, const float* __restrict__ res,
                     float* __restrict__ out, int M, int final_norm)
{
  int node = blockIdx.x * 8 + (threadIdx.x >> 5);
  int lane = threadIdx.x & 31;
  if (node >= M) return;
  float4 av = ((const float4*)(agg + (size_t)node * HID))[lane];
  float4 rv = ((const float4*)(res + (size_t)node * HID))[lane];
  float v[4] = {av.x + rv.x, av.y + rv.y, av.z + rv.z, av.w + rv.w};
  #pragma unroll
  for (int i = 0; i < 4; ++i) v[i] = v[i] > 0.f ? v[i] : (__expf(v[i]) - 1.f);
  if (final_norm) {
    float ss = v[0]*v[0] + v[1]*v[1] + v[2]*v[2] + v[3]*v[3];
    #pragma unroll
    for (int off = 16; off; off >>= 1) ss += __shfl_xor(ss, off, 32);
    float nrm = sqrtf(ss);
    nrm = nrm > 1e-12f ? nrm : 1e-12f;
    float inv = 1.0f / nrm;
    #pragma unroll
    for (int i = 0; i < 4; ++i) v[i] *= inv;
  }
  ((float4*)(out + (size_t)node * HID))[lane] = make_float4(v[0], v[1], v[2], v[3]);
}

extern "C" void kernel_launch(void* const* d_in, const int* in_sizes, int n_in,
                              void* d_out, int out_size, void* d_ws, size_t ws_size,
                              hipStream_t stream) {
  const float* x     = (const float*)d_in[0];
  const int*   ei    = (const int*)  d_in[1];   // [2, E] : row0=src, row1=dst
  const int*   etype = (const int*)  d_in[2];
  const float* l1W   = (const float*)d_in[3];
  const float* l1b   = (const float*)d_in[4];
  const float* l2W   = (const float*)d_in[5];
  const float* l2b   = (const float*)d_in[6];
  const float* W1    = (const float*)d_in[7];
  const float* Wr1   = (const float*)d_in[8];
  const float* a1    = (const float*)d_in[9];
  const float* Wres1 = (const float*)d_in[10];
  const float* rel1  = (const float*)d_in[11];
  const float* W2    = (const float*)d_in[12];
  const float* Wr2   = (const float*)d_in[13];
  const float* a2    = (const float*)d_in[14];
  const float* Wres2 = (const float*)d_in[15];
  const float* rel2  = (const float*)d_in[16];

  const int M    = in_sizes[0] / 768;
  const int E    = in_sizes[2];
  const int nRel = in_sizes[11] / RELD;

  // workspace carve-up (fits in L2: ~110 MB of hot intermediates)
  float*    ws     = (float*)d_ws;
  size_t    nf     = (size_t)M * HID;
  float*    bufA   = ws;               // h0 / out1 / out2
  float*    bufWx  = bufA  + nf;
  float*    bufRes = bufWx + nf;
  float*    bufAgg = bufRes + nf;
  float*    sA     = bufAgg + nf;
  float*    sB     = sA + M;
  float*    denom  = sB + M;
  unsigned* mOrd   = (unsigned*)(denom + M);
  float*    score  = (float*)(mOrd + M);
  float*    ee     = score + E;
  float*    sC     = ee + E;

  dim3 blk(256);
  const int gGemm = (M + 31) / 32;
  const int gNode = (M + 7) / 8;
  const int gEdge = (E + 255) / 256;
  const int gScat = (E + 7) / 8;
  const int gInit = (int)(((size_t)M * HID + 255) / 256);

  // h0 = leaky_relu(x @ l1_W^T + l1_b, 0.01)          [K=768]
  gemm_wmma_kernel<<<gGemm, blk, 0, stream>>>(x, l1W, l1b, bufA, M, 768, 1);

  // -------- layer 1 --------
  gemm_wmma_kernel<<<gGemm, blk, 0, stream>>>(bufA, W1,    nullptr, bufWx,  M, HID, 0);
  gemm_wmma_kernel<<<gGemm, blk, 0, stream>>>(bufA, Wres1, nullptr, bufRes, M, HID, 0);
  node_dots_kernel<<<gNode, blk, 0, stream>>>(bufWx, a1, sA, sB, M);
  relscore_kernel<<<1, 256, 0, stream>>>(Wr1, a1 + 2 * HID, rel1, sC, nRel);
  init_kernel<<<gInit, blk, 0, stream>>>(bufAgg, mOrd, denom, M);
  edge_score_kernel<<<gEdge, blk, 0, stream>>>(ei, ei + E, etype, sA, sB, sC, score, mOrd, E);
  edge_exp_kernel<<<gEdge, blk, 0, stream>>>(ei + E, score, mOrd, ee, denom, E);
  edge_scatter_kernel<<<gScat, blk, 0, stream>>>(ei, ei + E, bufWx, ee, denom, bufAgg, E);
  node_out_kernel<<<gNode, blk, 0, stream>>>(bufAgg, bufRes, bufA, M, 0);   // out1 -> bufA

  // -------- layer 2 --------
  gemm_wmma_kernel<<<gGemm, blk, 0, stream>>>(bufA, W2,    nullptr, bufWx,  M, HID, 0);
  gemm_wmma_kernel<<<gGemm, blk, 0, stream>>>(bufA, Wres2, nullptr, bufRes, M, HID, 0);
  node_dots_kernel<<<gNode, blk, 0, stream>>>(bufWx, a2, sA, sB, M);
  relscore_kernel<<<1, 256, 0, stream>>>(Wr2, a2 + 2 * HID, rel2, sC, nRel);
  init_kernel<<<gInit, blk, 0, stream>>>(bufAgg, mOrd, denom, M);
  edge_score_kernel<<<gEdge, blk, 0, stream>>>(ei, ei + E, etype, sA, sB, sC, score, mOrd, E);
  edge_exp_kernel<<<gEdge, blk, 0, stream>>>(ei + E, score, mOrd, ee, denom, E);
  edge_scatter_kernel<<<gScat, blk, 0, stream>>>(ei, ei + E, bufWx, ee, denom, bufAgg, E);
  node_out_kernel<<<gNode, blk, 0, stream>>>(bufAgg, bufRes, bufA, M, 1);   // out2 (L2-normed) -> bufA

  // out = leaky_relu(out2 @ l2_W^T + l2_b, 0.01)      [K=128]
  gemm_wmma_kernel<<<gGemm, blk, 0, stream>>>(bufA, l2W, l2b, (float*)d_out, M, HID, 1);
}